// PPR_66348654788876
// MI455X (gfx1250) — compile-verified
//
#include <hip/hip_runtime.h>
#include <hip/hip_bf16.h>

// PPR dense diffusion on MI455X (gfx1250).
//   P = theta1*(A + A^2 + A^3) + alpha*I,  theta1 = alpha*(1-alpha)
// Fused as:  B = A + A@A ;  P = theta1*(A + A@B) + alpha*I
// fp32 WMMA (V_WMMA_F32_16X16X4_F32) GEMMs, double-buffered LDS staging via
// GLOBAL_LOAD_ASYNC_TO_LDS_B128 (ASYNCcnt) when the builtin is available.
//
// Workspace layout (assumes ws_size >= 512MB + 64KB):
//   [0)                : A    (8192*8192 f32, 256MB)
//   [+N*N)             : Bmat (8192*8192 f32, 256MB)
//   [+2*N*N)           : deg  (8192 f32)
//   [+2*N*N+8192)      : dinv (8192 f32)

#define NMAT 8192
#define NMASK 8191
#define ALPHA_C 0.4f
#define THETA1_C 0.24f   // 0.4 * 0.6

typedef __attribute__((ext_vector_type(2))) float v2f;
typedef __attribute__((ext_vector_type(8))) float v8f;

// async-LDS builtin pointer types (signature leaked by the round-2 diagnostic:
// param0 is int __vector(4) in AS1 (__device__), param1 is the LDS side)
typedef int i4v __attribute__((vector_size(16)));
typedef __attribute__((address_space(1))) i4v as1_i4v;
typedef __attribute__((address_space(3))) i4v as3_i4v;

// ---------------- edge preprocessing ----------------

__global__ void ppr_deg_kernel(const int* __restrict__ col,
                               float* __restrict__ deg, int E) {
    int e = blockIdx.x * blockDim.x + threadIdx.x;
    if (e < E) {
        atomicAdd(&deg[col[e] & NMASK], 1.0f);
    }
}

__global__ void ppr_dinv_kernel(const float* __restrict__ deg,
                                float* __restrict__ dinv) {
    int i = blockIdx.x * blockDim.x + threadIdx.x;
    if (i < NMAT) {
        dinv[i] = rsqrtf(deg[i]);   // deg==0 -> +inf, matching pow(deg,-0.5)
    }
}

__global__ void ppr_scatter_kernel(const int* __restrict__ row,
                                   const int* __restrict__ col,
                                   const float* __restrict__ w,
                                   const float* __restrict__ dinv,
                                   float* __restrict__ A, int E) {
    int e = blockIdx.x * blockDim.x + threadIdx.x;
    if (e < E) {
        int r = row[e] & NMASK;
        int c = col[e] & NMASK;
        float v = dinv[r] * w[e] * dinv[c];
        atomicAdd(&A[(size_t)r * NMAT + c], v);
    }
}

// ---------------- async global->LDS helpers ----------------

#if defined(__gfx1250__) && __has_builtin(__builtin_amdgcn_global_load_async_to_lds_b128)
#define USE_ASYNC_LDS 1
#else
#define USE_ASYNC_LDS 0
#endif

__device__ __forceinline__ void copy16_g2l(const float* g, float* l) {
#if USE_ASYNC_LDS
    __builtin_amdgcn_global_load_async_to_lds_b128(
        (as1_i4v*)g, (as3_i4v*)l, 0, 0);
#else
    *reinterpret_cast<float4*>(l) = *reinterpret_cast<const float4*>(g);
#endif
}

__device__ __forceinline__ void wait_async_copies() {
#if USE_ASYNC_LDS
#if __has_builtin(__builtin_amdgcn_s_wait_asynccnt)
    __builtin_amdgcn_s_wait_asynccnt(0);
#else
    asm volatile("s_wait_asynccnt 0x0" ::: "memory");
#endif
#endif
}

// ---------------- fp32 WMMA GEMM ----------------
// C = accScale*(A@B) + epScale*Aep + diagVal*I
// block tile 128x128, 256 threads (8 waves), wave tile 32x64 (2x4 WMMA tiles)

#define BM 128
#define BN 128
#define BK 32
#define SA 36     // LDS stride of A tile rows (m-major): b128-aligned, no bank conflicts
#define SB 136    // LDS stride of B tile rows (k-major): half-waves hit disjoint banks

__global__ __launch_bounds__(256)
void ppr_gemm_wmma_f32(const float* __restrict__ A,
                       const float* __restrict__ B,
                       const float* __restrict__ Aep,
                       float* __restrict__ C,
                       float accScale, float epScale, float diagVal) {
    __shared__ float As[2][BM * SA];   // m-major: As[buf][m*SA + k]
    __shared__ float Bs[2][BK * SB];   // k-major: Bs[buf][k*SB + n]

    const int tid  = threadIdx.x;
    const int lane = tid & 31;
    const int wid  = tid >> 5;          // 0..7
    const int mW   = (wid >> 1) * 32;   // wave row origin (4 waves of 32 rows)
    const int nW   = (wid & 1) * 64;    // wave col origin (2 waves of 64 cols)

    const int m0 = blockIdx.y * BM;
    const int n0 = blockIdx.x * BN;

    const int half = lane >> 4;         // 0 or 1
    const int lm   = lane & 15;

    v8f acc[2][4] = {};

    // ---- stage one K-chunk (contiguous b128 copies, async when available) ----
    auto stage = [&](int k0, int buf) {
        float* Asb = &As[buf][0];
        float* Bsb = &Bs[buf][0];
        #pragma unroll
        for (int t4 = 0; t4 < 4; ++t4) {
            int idx = tid + t4 * 256;           // 0..1023
            int r   = idx >> 3;                 // 0..127
            int kq  = (idx & 7) << 2;           // 0,4,...,28
            copy16_g2l(&A[(size_t)(m0 + r) * NMAT + k0 + kq], &Asb[r * SA + kq]);
        }
        #pragma unroll
        for (int t4 = 0; t4 < 4; ++t4) {
            int idx = tid + t4 * 256;           // 0..1023
            int kr  = idx >> 5;                 // 0..31
            int nq  = (idx & 31) << 2;          // 0..124
            copy16_g2l(&B[(size_t)(k0 + kr) * NMAT + n0 + nq], &Bsb[kr * SB + nq]);
        }
    };

    const int nk = NMAT / BK;           // 256 K-chunks
    stage(0, 0);
    wait_async_copies();
    __syncthreads();

    for (int t = 0; t < nk; ++t) {
        const int cur = t & 1;
        if (t + 1 < nk) {
            stage((t + 1) * BK, cur ^ 1);   // prefetch next chunk (other buffer)
        }

        const float* Asb = &As[cur][0];
        const float* Bsb = &Bs[cur][0];

        #pragma unroll
        for (int kk = 0; kk < BK; kk += 4) {
            const int kb = kk + 2 * half;       // this lane's consecutive K pair
            v2f af[2];
            #pragma unroll
            for (int mt = 0; mt < 2; ++mt) {
                int m = mW + mt * 16 + lm;
                af[mt] = *reinterpret_cast<const v2f*>(&Asb[m * SA + kb]);  // ds_load_b64
            }
            v2f bf[4];
            #pragma unroll
            for (int nt = 0; nt < 4; ++nt) {
                int n = nW + nt * 16 + lm;
                bf[nt] = (v2f){ Bsb[(kb + 0) * SB + n],
                                Bsb[(kb + 1) * SB + n] };
            }
            #pragma unroll
            for (int mt = 0; mt < 2; ++mt) {
                #pragma unroll
                for (int nt = 0; nt < 4; ++nt) {
                    acc[mt][nt] = __builtin_amdgcn_wmma_f32_16x16x4_f32(
                        false, af[mt], false, bf[nt],
                        (short)0, acc[mt][nt], false, false);
                }
            }
        }

        wait_async_copies();    // prefetch landed (overlapped with the WMMAs above)
        __syncthreads();
    }

    // ---- fused epilogue: C = accScale*acc + epScale*Aep + diagVal*I ----
    #pragma unroll
    for (int mt = 0; mt < 2; ++mt) {
        #pragma unroll
        for (int nt = 0; nt < 4; ++nt) {
            #pragma unroll
            for (int v = 0; v < 8; ++v) {
                int gm = m0 + mW + mt * 16 + v + 8 * half;
                int gn = n0 + nW + nt * 16 + lm;
                size_t off = (size_t)gm * NMAT + gn;
                float r = accScale * acc[mt][nt][v] + epScale * Aep[off];
                if (gm == gn) r += diagVal;
                C[off] = r;
            }
        }
    }
}

// ---------------- launch ----------------

extern "C" void kernel_launch(void* const* d_in, const int* in_sizes, int n_in,
                              void* d_out, int out_size, void* d_ws, size_t ws_size,
                              hipStream_t stream) {
    // inputs: x [N,64] f32 (unused: only shape matters), edge_attr [E] f32,
    //         edge_index [2,E] int (row then col)
    const float* edge_attr = (const float*)d_in[1];
    const int*   eidx      = (const int*)d_in[2];
    const int E = in_sizes[1];
    const int* row = eidx;
    const int* col = eidx + E;

    float* A    = (float*)d_ws;
    float* Bmat = A + (size_t)NMAT * NMAT;
    float* deg  = Bmat + (size_t)NMAT * NMAT;
    float* dinv = deg + NMAT;

    (void)hipMemsetAsync(A,   0, (size_t)NMAT * NMAT * sizeof(float), stream);
    (void)hipMemsetAsync(deg, 0, NMAT * sizeof(float), stream);

    ppr_deg_kernel<<<(E + 255) / 256, 256, 0, stream>>>(col, deg, E);
    ppr_dinv_kernel<<<NMAT / 256, 256, 0, stream>>>(deg, dinv);
    ppr_scatter_kernel<<<(E + 255) / 256, 256, 0, stream>>>(row, col, edge_attr,
                                                            dinv, A, E);

    dim3 grid(NMAT / BN, NMAT / BM);
    dim3 block(256);
    // Bmat = A + A@A
    ppr_gemm_wmma_f32<<<grid, block, 0, stream>>>(A, A, A, Bmat,
                                                  1.0f, 1.0f, 0.0f);
    // P = theta1*(A + A@Bmat) + alpha*I
    ppr_gemm_wmma_f32<<<grid, block, 0, stream>>>(A, Bmat, A, (float*)d_out,
                                                  THETA1_C, THETA1_C, ALPHA_C);
}